// RecurrentGCN_858993459512
// MI455X (gfx1250) — compile-verified
//
#include <hip/hip_runtime.h>
#include <hip/hip_bf16.h>
#include <math.h>

// GCLSTM (ChebConv K=1) fused cell for MI455X / gfx1250.
// Bandwidth-bound: ~512MB traffic -> ~22us floor @ 23.3 TB/s.
// Matrix math on V_WMMA_F32_16X16X4_F32 (exact fp32, matches reference).
//
// Fragment layouts (ISA 7.12.2, 32-bit):
//   A 16x4 : lane%16 = M row, (lane/16, vgpr) = K pair -> per-lane b64 from x/h row
//   B 4x16 : lane%16 = N col, (lane/16, vgpr) = K pair -> per-lane b64 from W^T in LDS
//            (+4-float row pad => the two b64s of a ds_load_2addr hit all 64 banks once)
//   D 16x16: lane%16 = channel, vgpr v = node v + 8*(lane/16) -> gates map to acc pairs

typedef __attribute__((ext_vector_type(2))) float v2f;
typedef __attribute__((ext_vector_type(8))) float v8f;

#define IN_CH 128
#define OUT_CH 32
#define WAVES 8
#define WXT_STRIDE 132   // 128 + 4 pad
#define THT_STRIDE 36    // 32 + 4 pad

__device__ __forceinline__ float sigmoid_(float v) {
    return 1.0f / (1.0f + __expf(-v));   // saturates cleanly at both ends, no NaN
}
__device__ __forceinline__ float fast_tanh_(float v) {
    // clamp keeps __expf finite; tanh is fully saturated well before |15|
    float xs = fminf(fmaxf(v, -15.0f), 15.0f);
    float e  = __expf(2.0f * xs);
    return (e - 1.0f) / (e + 1.0f);
}

__launch_bounds__(256, 1)
__global__ void gclstm_wmma_kernel(const float* __restrict__ x,
                                   const float* __restrict__ h,
                                   const float* __restrict__ c,
                                   const float* __restrict__ Wx,     // [4][128][32]
                                   const float* __restrict__ Theta,  // [4][32][32]
                                   const float* __restrict__ conv_b, // [4][32]
                                   const float* __restrict__ w_c,    // [3][32]
                                   const float* __restrict__ bb,     // [4][32]
                                   float* __restrict__ h_out,
                                   float* __restrict__ c_out,
                                   int n)
{
    // Weights transposed to out-major [n_col][k] so B fragments are per-lane b64s.
    __shared__ float sWxT[128 * WXT_STRIDE]; // 67,584 B
    __shared__ float sThT[128 * THT_STRIDE]; // 18,432 B

    const int tid = threadIdx.x;
    for (int idx = tid; idx < 4 * IN_CH * OUT_CH; idx += blockDim.x) {
        int g   = idx / (IN_CH * OUT_CH);
        int rem = idx % (IN_CH * OUT_CH);
        int f   = rem / OUT_CH;
        int o   = rem % OUT_CH;
        sWxT[(g * OUT_CH + o) * WXT_STRIDE + f] = Wx[idx];
    }
    for (int idx = tid; idx < 4 * OUT_CH * OUT_CH; idx += blockDim.x) {
        int g   = idx / (OUT_CH * OUT_CH);
        int rem = idx % (OUT_CH * OUT_CH);
        int i   = rem / OUT_CH;
        int o   = rem % OUT_CH;
        sThT[(g * OUT_CH + o) * THT_STRIDE + i] = Theta[idx];
    }
    __syncthreads();

    const int lane = tid & 31;
    const int wave = tid >> 5;
    const int lrow = lane & 15;  // A: M row | B/D: N column (channel)
    const int lhi  = lane >> 4;  // selects K pair (A/B) / M+8 (D)

    const int chan_lo = lrow;
    const int chan_hi = lrow + 16;

    // Per-lane gate biases (conv_b + b) for each accumulator j = 2*gate + half.
    float bias[8], wci[2], wcf[2], wco[2];
    #pragma unroll
    for (int g = 0; g < 4; ++g) {
        bias[2 * g + 0] = conv_b[g * 32 + chan_lo] + bb[g * 32 + chan_lo];
        bias[2 * g + 1] = conv_b[g * 32 + chan_hi] + bb[g * 32 + chan_hi];
    }
    wci[0] = w_c[0 * 32 + chan_lo]; wci[1] = w_c[0 * 32 + chan_hi];
    wcf[0] = w_c[1 * 32 + chan_lo]; wcf[1] = w_c[1 * 32 + chan_hi];
    wco[0] = w_c[2 * 32 + chan_lo]; wco[1] = w_c[2 * 32 + chan_hi];

    const int ntiles = (n + 15) / 16;
    const int tstep  = gridDim.x * WAVES;
    for (int t = blockIdx.x * WAVES + wave; t < ntiles; t += tstep) {
        const int base = t * 16;
        int arow = base + lrow;
        if (arow > n - 1) arow = n - 1;   // clamp loads; WMMA requires EXEC all-1
        const float* xrow = x + (size_t)arow * IN_CH;
        const float* hrow = h + (size_t)arow * OUT_CH;

        // Prefetch next grid-stride tile's A-rows (dominant HBM stream).
        {
            int tn = t + tstep;
            if (tn < ntiles) {
                const float* xn = x + (size_t)(tn * 16 + lrow) * IN_CH;
                __builtin_prefetch(xn, 0, 3);
                __builtin_prefetch(xn + 64, 0, 3);
            }
        }

        // acc[2g+half]: gate g, channels lo/hi; D element v -> node base+v+8*lhi
        v8f acc[8];
        #pragma unroll
        for (int j = 0; j < 8; ++j) {
            v8f z;
            #pragma unroll
            for (int v = 0; v < 8; ++v) z[v] = bias[j];
            acc[j] = z;
        }

        // ---- x @ W_x : K = 128, 32 K-steps x 8 N-tiles ----
        #pragma unroll 2
        for (int ks = 0; ks < IN_CH / 4; ++ks) {
            const int k0 = ks * 4 + 2 * lhi;
            v2f a = *(const v2f*)(xrow + k0);
            #pragma unroll
            for (int nt = 0; nt < 8; ++nt) {
                v2f bf = *(const v2f*)(&sWxT[(nt * 16 + lrow) * WXT_STRIDE + k0]);
                acc[nt] = __builtin_amdgcn_wmma_f32_16x16x4_f32(
                    false, a, false, bf, (short)0, acc[nt], false, false);
            }
        }
        // ---- h @ Theta : K = 32, 8 K-steps x 8 N-tiles ----
        #pragma unroll 2
        for (int ks = 0; ks < OUT_CH / 4; ++ks) {
            const int k0 = ks * 4 + 2 * lhi;
            v2f a = *(const v2f*)(hrow + k0);
            #pragma unroll
            for (int nt = 0; nt < 8; ++nt) {
                v2f bf = *(const v2f*)(&sThT[(nt * 16 + lrow) * THT_STRIDE + k0]);
                acc[nt] = __builtin_amdgcn_wmma_f32_16x16x4_f32(
                    false, a, false, bf, (short)0, acc[nt], false, false);
            }
        }

        // ---- LSTM elementwise on D fragments ----
        #pragma unroll
        for (int v = 0; v < 8; ++v) {
            const int mrow = base + v + 8 * lhi;
            const int mc   = (mrow < n) ? mrow : (n - 1);
            const size_t ro = (size_t)mc * OUT_CH;
            #pragma unroll
            for (int half = 0; half < 2; ++half) {
                const int chan = lrow + 16 * half;
                const float cv = c[ro + chan];
                const float gi = acc[0 + half][v] + (half ? wci[1] : wci[0]) * cv;
                const float gf = acc[2 + half][v] + (half ? wcf[1] : wcf[0]) * cv;
                const float gc = acc[4 + half][v];
                const float go = acc[6 + half][v];
                const float iv = sigmoid_(gi);
                const float fv = sigmoid_(gf);
                const float cn = fv * cv + iv * fast_tanh_(gc);
                const float ov = sigmoid_(go + (half ? wco[1] : wco[0]) * cn);
                const float hn = ov * fast_tanh_(cn);
                if (mrow < n) {
                    h_out[ro + chan] = hn;
                    c_out[ro + chan] = cn;
                }
            }
        }
    }
}

extern "C" void kernel_launch(void* const* d_in, const int* in_sizes, int n_in,
                              void* d_out, int out_size, void* d_ws, size_t ws_size,
                              hipStream_t stream) {
    // setup_inputs order: x, edge_index, edge_weight, h, c, W_x, Theta, conv_b, w_c, b
    const float* x      = (const float*)d_in[0];
    const float* h      = (const float*)d_in[3];
    const float* c      = (const float*)d_in[4];
    const float* Wx     = (const float*)d_in[5];
    const float* Theta  = (const float*)d_in[6];
    const float* conv_b = (const float*)d_in[7];
    const float* w_c    = (const float*)d_in[8];
    const float* b      = (const float*)d_in[9];
    const int n = in_sizes[0] / IN_CH;

    float* h_out = (float*)d_out;                       // (h_new, c_new) concat
    float* c_out = h_out + (size_t)n * OUT_CH;

    const int ntiles = (n + 15) / 16;
    int blocks = (ntiles + WAVES - 1) / WAVES;
    if (blocks > 2048) blocks = 2048;                   // grid-stride amortizes LDS fill

    gclstm_wmma_kernel<<<blocks, 256, 0, stream>>>(
        x, h, c, Wx, Theta, conv_b, w_c, b, h_out, c_out, n);
}